// LuongAttnDecoderRNN_21603685498896
// MI455X (gfx1250) — compile-verified
//
#include <hip/hip_runtime.h>
#include <math.h>

// Problem sizes (fixed by the reference)
#define B_  64
#define E_  512
#define H_  1024
#define V_  50257
#define S_  2048
#define P_  32            // split-S partials per batch row (fused attention)

typedef float v8f __attribute__((ext_vector_type(8)));
typedef float v2f __attribute__((ext_vector_type(2)));
typedef int   v4i __attribute__((ext_vector_type(4)));

#define LDS_AS __attribute__((address_space(3)))

#if defined(__has_builtin)
#if __has_builtin(__builtin_amdgcn_global_load_async_to_lds_b128)
#define USE_ASYNC_LDS 1
#endif
#endif

// =====================================================================
// WMMA f32 GEMM:  C[64,N] = A[64,K] @ W[N,K]^T  (+bias, optional tanh)
//   - block = 256 threads = 8 waves; block tile = 64 (all of M) x 128 (N)
//   - K tiled by 32, staged in LDS (padded stride 33 -> conflict-free)
//   - W tile staged via GLOBAL_LOAD_ASYNC_TO_LDS_B128 (ASYNCcnt) when the
//     toolchain exposes the builtin; synchronous float4 path otherwise
// Fragment layouts per CDNA5 ISA 7.12.2 (f32, 16x16x4):
//   A 16x4 : lane&15 = M row; VGPR0 = K {0 | 2}, VGPR1 = K {1 | 3}
//   B 4x16 : lane&15 = N col; same K striping across VGPR x wave-half
//   C/D    : VGPR v -> M = v + 8*(lane>>4), N = lane&15
// =====================================================================
#define BN 128
#define BK 32

__global__ __launch_bounds__(256)
void gemm_wmma_f32(const float* __restrict__ A, const float* __restrict__ W,
                   const float* __restrict__ bias, float* __restrict__ C,
                   int N, int K, int act /*0=none,1=tanh*/)
{
  __shared__ float sA[64][BK + 1];
  __shared__ float sW[BN][BK + 1];

  const int tid  = threadIdx.x;
  const int wave = tid >> 5;
  const int lane = tid & 31;
  const int half = lane >> 4;    // wave half: 0 -> K base 0, 1 -> K base 2
  const int q    = lane & 15;    // M row (A frag) / N col (B frag)
  const int n0   = blockIdx.x * BN;

  v8f acc[4] = {};

  for (int k0 = 0; k0 < K; k0 += BK) {
    // ---- stage A tile: 64x32 floats = 512 float4, 2 per thread (coalesced)
    #pragma unroll
    for (int i = 0; i < 2; ++i) {
      int idx = tid + i * 256;
      int r   = idx >> 3;
      int c4  = idx & 7;
      const float4 v = *(const float4*)(A + (size_t)r * K + k0 + c4 * 4);
      sA[r][c4*4+0] = v.x; sA[r][c4*4+1] = v.y;
      sA[r][c4*4+2] = v.z; sA[r][c4*4+3] = v.w;
    }
    // ---- stage W tile: 128x32 floats, 4 x 16B per thread; zero-pad rows >= N
#ifdef USE_ASYNC_LDS
    #pragma unroll
    for (int i = 0; i < 4; ++i) {
      int idx = tid + i * 256;
      int r   = idx >> 3;
      int c4  = idx & 7;
      int g   = n0 + r;
      if (g < N) {
        __builtin_amdgcn_global_load_async_to_lds_b128(
            (v4i*)(W + (size_t)g * K + k0 + c4 * 4),
            (LDS_AS v4i*)&sW[r][c4 * 4], 0, 0);
      } else {
        sW[r][c4*4+0] = 0.f; sW[r][c4*4+1] = 0.f;
        sW[r][c4*4+2] = 0.f; sW[r][c4*4+3] = 0.f;
      }
    }
    asm volatile("s_wait_asynccnt 0x0" ::: "memory");
#else
    #pragma unroll
    for (int i = 0; i < 4; ++i) {
      int idx = tid + i * 256;
      int r   = idx >> 3;
      int c4  = idx & 7;
      int g   = n0 + r;
      float4 v = make_float4(0.f, 0.f, 0.f, 0.f);
      if (g < N) {
        v = *(const float4*)(W + (size_t)g * K + k0 + c4 * 4);
        if (k0 + BK < K)
          __builtin_prefetch(W + (size_t)g * K + k0 + BK + c4 * 4, 0, 0);
      }
      sW[r][c4*4+0] = v.x; sW[r][c4*4+1] = v.y;
      sW[r][c4*4+2] = v.z; sW[r][c4*4+3] = v.w;
    }
#endif
    __syncthreads();

    #pragma unroll
    for (int kk = 0; kk < BK; kk += 4) {
      v2f b;
      b.x = sW[(wave << 4) + q][kk + 2*half];
      b.y = sW[(wave << 4) + q][kk + 2*half + 1];
      #pragma unroll
      for (int mt = 0; mt < 4; ++mt) {
        v2f a;
        a.x = sA[(mt << 4) + q][kk + 2*half];
        a.y = sA[(mt << 4) + q][kk + 2*half + 1];
        acc[mt] = __builtin_amdgcn_wmma_f32_16x16x4_f32(
            false, a, false, b, (short)0, acc[mt], false, false);
      }
    }
    __syncthreads();
  }

  // ---- epilogue: bias + optional tanh, store (bounds-checked in N)
  const int col = n0 + (wave << 4) + q;
  if (col < N) {
    const float bval = bias ? bias[col] : 0.0f;
    #pragma unroll
    for (int mt = 0; mt < 4; ++mt) {
      #pragma unroll
      for (int v = 0; v < 8; ++v) {
        int   row = (mt << 4) + v + (half << 3);
        float val = acc[mt][v] + bval;
        if (act == 1) val = tanhf(val);
        C[(size_t)row * N + col] = val;
      }
    }
  }
}

// =====================================================================
// GRU elementwise combine (PyTorch gate order r,z,n; biases already in gi/gh)
// =====================================================================
__global__ __launch_bounds__(256)
void gru_combine(const float* __restrict__ gi, const float* __restrict__ gh,
                 const float* __restrict__ h_old, float* __restrict__ h_out,
                 float* __restrict__ concat_in)
{
  const int idx = blockIdx.x * 256 + threadIdx.x;   // [0, B*H)
  const int b = idx / H_;
  const int j = idx - b * H_;
  const float* gib = gi + (size_t)b * 3 * H_;
  const float* ghb = gh + (size_t)b * 3 * H_;
  const float r = 1.f / (1.f + __expf(-(gib[j]        + ghb[j])));
  const float z = 1.f / (1.f + __expf(-(gib[j + H_]   + ghb[j + H_])));
  const float n = tanhf(gib[j + 2*H_] + r * ghb[j + 2*H_]);
  const float h = (1.f - z) * n + z * h_old[idx];
  h_out[idx] = h;                                   // -> "hidden" output
  concat_in[(size_t)b * 2 * H_ + j] = h;            // first half of concat
}

// =====================================================================
// FUSED attention pass 1 (flash-style): single traversal of enc.
// One wave owns partial p for batch row b; processes S/P_ enc rows.
// Per-lane h ownership: h = 128*j + 4*lane + k (float4 j=0..7, k=0..3).
// Keeps h_new, running context, and the streamed enc row in registers:
// enc is read from HBM exactly once (512 MB instead of 1 GB).
// Raw energies are stored so exact softmax weights can be emitted later.
// =====================================================================
__global__ __launch_bounds__(256)
void attn_pass1(const float* __restrict__ hnew, const float* __restrict__ enc,
                float* __restrict__ energies, float* __restrict__ pm,
                float* __restrict__ pl, float* __restrict__ pctx)
{
  const int b    = blockIdx.y;
  const int wave = threadIdx.x >> 5;
  const int lane = threadIdx.x & 31;
  const int p    = blockIdx.x * 8 + wave;           // gridDim.x = P_/8

  float4 hreg[8], ctx[8];
  const float4* hrow = (const float4*)(hnew + (size_t)b * H_);
  #pragma unroll
  for (int j = 0; j < 8; ++j) {
    hreg[j] = hrow[j * 32 + lane];
    ctx[j]  = make_float4(0.f, 0.f, 0.f, 0.f);
  }

  float m = -3.4e38f, l = 0.f;

  for (int t = 0; t < S_ / P_; ++t) {
    const int s = p + P_ * t;
    const float4* erow = (const float4*)(enc + ((size_t)s * B_ + b) * H_);
    float4 row[8];
    float  d = 0.f;
    #pragma unroll
    for (int j = 0; j < 8; ++j) {
      row[j] = erow[j * 32 + lane];
      d += row[j].x * hreg[j].x + row[j].y * hreg[j].y
         + row[j].z * hreg[j].z + row[j].w * hreg[j].w;
    }
    #pragma unroll
    for (int o = 16; o >= 1; o >>= 1) d += __shfl_xor(d, o, 32);

    // online softmax update (uniform across the wave)
    const float mnew  = fmaxf(m, d);
    const float scale = __expf(m - mnew);
    const float w     = __expf(d - mnew);
    l = l * scale + w;
    #pragma unroll
    for (int j = 0; j < 8; ++j) {
      ctx[j].x = ctx[j].x * scale + w * row[j].x;
      ctx[j].y = ctx[j].y * scale + w * row[j].y;
      ctx[j].z = ctx[j].z * scale + w * row[j].z;
      ctx[j].w = ctx[j].w * scale + w * row[j].w;
    }
    m = mnew;
    if (lane == 0) energies[(size_t)b * S_ + s] = d;
  }

  if (lane == 0) { pm[b * P_ + p] = m; pl[b * P_ + p] = l; }
  float4* outc = (float4*)(pctx + ((size_t)b * P_ + p) * H_);
  #pragma unroll
  for (int j = 0; j < 8; ++j) outc[j * 32 + lane] = ctx[j];
}

// =====================================================================
// Combine split-S partials: m = max_p m_p ; l = sum_p l_p*exp(m_p-m);
// context[b,h] = (1/l) * sum_p exp(m_p-m) * ctx_p[h]  -> concat 2nd half
// Block (blockIdx.x==0) also publishes (m,l) for the weights kernel.
// =====================================================================
__global__ __launch_bounds__(256)
void attn_combine(const float* __restrict__ pm, const float* __restrict__ pl,
                  const float* __restrict__ pctx, float* __restrict__ concat_in,
                  float* __restrict__ mfin, float* __restrict__ lfin)
{
  __shared__ float coef[P_];
  __shared__ float linv_s;
  const int b = blockIdx.y, tid = threadIdx.x;

  if (tid < P_) {
    const float mp = pm[b * P_ + tid];
    float mx = mp;
    #pragma unroll
    for (int o = 16; o >= 1; o >>= 1) mx = fmaxf(mx, __shfl_xor(mx, o, 32));
    const float c = __expf(mp - mx);
    float ls = pl[b * P_ + tid] * c;
    #pragma unroll
    for (int o = 16; o >= 1; o >>= 1) ls += __shfl_xor(ls, o, 32);
    coef[tid] = c;
    if (tid == 0) {
      linv_s = 1.f / ls;
      if (blockIdx.x == 0) { mfin[b] = mx; lfin[b] = ls; }
    }
  }
  __syncthreads();

  const float linv = linv_s;
  const int h = blockIdx.x * 256 + tid;
  float acc = 0.f;
  #pragma unroll 4
  for (int p = 0; p < P_; ++p)
    acc += coef[p] * pctx[((size_t)b * P_ + p) * H_ + h];
  concat_in[(size_t)b * 2 * H_ + H_ + h] = acc * linv;
}

// attn[b,s] = exp(e[b,s] - m_b) / l_b  (exactly the reference softmax)
__global__ __launch_bounds__(256)
void attn_weights_k(const float* __restrict__ energies,
                    const float* __restrict__ mfin, const float* __restrict__ lfin,
                    float* __restrict__ attn)
{
  const int b = blockIdx.y;
  const int s = blockIdx.x * 256 + threadIdx.x;
  attn[(size_t)b * S_ + s] =
      __expf(energies[(size_t)b * S_ + s] - mfin[b]) * (1.f / lfin[b]);
}

// =====================================================================
// Fallback two-pass attention (used only if workspace is too small)
// =====================================================================
__global__ __launch_bounds__(256)
void attn_energies(const float* __restrict__ hnew, const float* __restrict__ enc,
                   float* __restrict__ energies)
{
  __shared__ float hrow[H_];
  const int b   = blockIdx.y;
  const int tid = threadIdx.x;
  for (int i = tid; i < H_; i += 256) hrow[i] = hnew[(size_t)b * H_ + i];
  __syncthreads();
  const int wave = tid >> 5, lane = tid & 31;
  const int s    = blockIdx.x * 8 + wave;
  const float* erow = enc + ((size_t)s * B_ + b) * H_;
  float acc = 0.f;
  #pragma unroll 4
  for (int h = lane; h < H_; h += 32) acc += hrow[h] * erow[h];
  #pragma unroll
  for (int m = 16; m >= 1; m >>= 1) acc += __shfl_xor(acc, m, 32);
  if (lane == 0) energies[(size_t)b * S_ + s] = acc;
}

__global__ __launch_bounds__(256)
void softmax_rows(const float* __restrict__ energies, float* __restrict__ attn)
{
  __shared__ float red[256];
  const int b = blockIdx.x, tid = threadIdx.x;
  const float* e = energies + (size_t)b * S_;
  float* a = attn + (size_t)b * S_;

  float m = -3.4e38f;
  for (int s = tid; s < S_; s += 256) m = fmaxf(m, e[s]);
  red[tid] = m; __syncthreads();
  for (int off = 128; off >= 1; off >>= 1) {
    if (tid < off) red[tid] = fmaxf(red[tid], red[tid + off]);
    __syncthreads();
  }
  m = red[0]; __syncthreads();

  float sum = 0.f;
  for (int s = tid; s < S_; s += 256) { float v = __expf(e[s] - m); a[s] = v; sum += v; }
  red[tid] = sum; __syncthreads();
  for (int off = 128; off >= 1; off >>= 1) {
    if (tid < off) red[tid] += red[tid + off];
    __syncthreads();
  }
  const float inv = 1.f / red[0];
  for (int s = tid; s < S_; s += 256) a[s] *= inv;
}

__global__ __launch_bounds__(256)
void attn_context(const float* __restrict__ attn, const float* __restrict__ enc,
                  float* __restrict__ concat_in)
{
  __shared__ float arow[S_];
  const int b = blockIdx.y, tid = threadIdx.x;
  for (int i = tid; i < S_; i += 256) arow[i] = attn[(size_t)b * S_ + i];
  __syncthreads();
  const int h = blockIdx.x * 256 + tid;
  float acc = 0.f;
  #pragma unroll 4
  for (int s = 0; s < S_; ++s)
    acc += arow[s] * enc[((size_t)s * B_ + b) * H_ + h];
  concat_in[(size_t)b * 2 * H_ + H_ + h] = acc;
}

// =====================================================================
// Launch
// =====================================================================
extern "C" void kernel_launch(void* const* d_in, const int* in_sizes, int n_in,
                              void* d_out, int out_size, void* d_ws, size_t ws_size,
                              hipStream_t stream)
{
  (void)in_sizes; (void)n_in; (void)out_size;

  const float* x    = (const float*)d_in[0];   // [1,B,E]
  const float* h0   = (const float*)d_in[1];   // [1,B,H]
  const float* enc  = (const float*)d_in[2];   // [S,B,H]
  const float* w_ih = (const float*)d_in[3];   // [3H,E]
  const float* w_hh = (const float*)d_in[4];   // [3H,H]
  const float* b_ih = (const float*)d_in[5];
  const float* b_hh = (const float*)d_in[6];
  const float* Wc   = (const float*)d_in[7];   // [H,2H]
  const float* bc   = (const float*)d_in[8];
  const float* Wo   = (const float*)d_in[9];   // [V,H]
  const float* bo   = (const float*)d_in[10];

  // Output tuple, flat in return order: output[B,V], hidden[1,B,H], attn[B,1,S]
  float* out    = (float*)d_out;
  float* hidden = out + (size_t)B_ * V_;
  float* attn   = hidden + (size_t)B_ * H_;

  // Workspace layout (floats)
  float* ws         = (float*)d_ws;
  float* gi         = ws;                                 // B*3H
  float* gh         = gi + (size_t)B_ * 3 * H_;           // B*3H
  float* energies   = gh + (size_t)B_ * 3 * H_;           // B*S
  float* concat_in  = energies + (size_t)B_ * S_;         // B*2H
  float* concat_out = concat_in + (size_t)B_ * 2 * H_;    // B*H
  float* pm         = concat_out + (size_t)B_ * H_;       // B*P
  float* pl         = pm + (size_t)B_ * P_;               // B*P
  float* mfin       = pl + (size_t)B_ * P_;               // B
  float* lfin       = mfin + B_;                          // B
  float* pctx       = lfin + B_;                          // B*P*H
  const size_t fused_need =
      (size_t)(pctx - ws + (size_t)B_ * P_ * H_) * sizeof(float);

  // 1) GRU input/hidden gate GEMMs (WMMA f32)
  gemm_wmma_f32<<<dim3((3*H_ + BN - 1) / BN), 256, 0, stream>>>(
      x,  w_ih, b_ih, gi, 3*H_, E_, 0);
  gemm_wmma_f32<<<dim3((3*H_ + BN - 1) / BN), 256, 0, stream>>>(
      h0, w_hh, b_hh, gh, 3*H_, H_, 0);

  // 2) GRU gate combine -> hidden output + first half of concat input
  gru_combine<<<dim3(B_ * H_ / 256), 256, 0, stream>>>(gi, gh, h0, hidden, concat_in);

  // 3-5) attention: fused single-pass over enc when workspace permits
  if (ws_size >= fused_need) {
    attn_pass1<<<dim3(P_ / 8, B_), 256, 0, stream>>>(
        hidden, enc, energies, pm, pl, pctx);
    attn_combine<<<dim3(H_ / 256, B_), 256, 0, stream>>>(
        pm, pl, pctx, concat_in, mfin, lfin);
    attn_weights_k<<<dim3(S_ / 256, B_), 256, 0, stream>>>(
        energies, mfin, lfin, attn);
  } else {
    attn_energies<<<dim3(S_ / 8, B_), 256, 0, stream>>>(hidden, enc, energies);
    softmax_rows<<<dim3(B_), 256, 0, stream>>>(energies, attn);
    attn_context<<<dim3(H_ / 256, B_), 256, 0, stream>>>(attn, enc, concat_in);
  }

  // 6) concat projection + tanh (WMMA f32)
  gemm_wmma_f32<<<dim3((H_ + BN - 1) / BN), 256, 0, stream>>>(
      concat_in, Wc, bc, concat_out, H_, 2*H_, 1);

  // 7) vocabulary projection (dominant GEMM; WMMA f32)
  gemm_wmma_f32<<<dim3((V_ + BN - 1) / BN), 256, 0, stream>>>(
      concat_out, Wo, bo, out, V_, H_, 0);
}